// incept_triplet_22213570855625
// MI455X (gfx1250) — compile-verified
//
#include <hip/hip_runtime.h>
#include <hip/hip_bf16.h>

#define BB 32
#define CC 256
#define NN 1024

typedef __attribute__((ext_vector_type(2))) float v2f;
typedef __attribute__((ext_vector_type(8))) float v8f;

// ---------------------------------------------------------------------------
// Kernel 1a: pool over C (channel axis) -> pool_t (B, 4, N)  [mean,std,max,min]
// One thread per (b, n); lanes have consecutive n -> coalesced per-c row reads.
// ---------------------------------------------------------------------------
__global__ void pool_over_C(const float* __restrict__ x, float* __restrict__ pool_t) {
    int gid = blockIdx.x * blockDim.x + threadIdx.x;      // 0 .. B*N-1
    int b = gid >> 10;                                    // / N
    int n = gid & (NN - 1);
    const float* xp = x + (size_t)b * (CC * NN) + n;
    float s = 0.f, sq = 0.f;
    float mx = -__builtin_inff(), mn = __builtin_inff();
    #pragma unroll 4
    for (int c = 0; c < CC; ++c) {
        float v = xp[(size_t)c * NN];
        s += v; sq += v * v;
        mx = fmaxf(mx, v); mn = fminf(mn, v);
    }
    float mean = s * (1.0f / CC);
    float var  = (sq - s * mean) * (1.0f / (CC - 1));     // unbiased (ddof=1)
    float sd   = sqrtf(fmaxf(var, 0.f));
    float* pt = pool_t + (size_t)b * (4 * NN) + n;
    pt[0 * NN] = mean; pt[1 * NN] = sd; pt[2 * NN] = mx; pt[3 * NN] = mn;
}

// ---------------------------------------------------------------------------
// Kernel 1b: pool over N (length axis) -> pool_c (B, 4, C)
// One 256-thread block per (b, c); coalesced contiguous reads + LDS tree.
// ---------------------------------------------------------------------------
__global__ void pool_over_N(const float* __restrict__ x, float* __restrict__ pool_c) {
    int b = blockIdx.x >> 8;                              // / C
    int c = blockIdx.x & (CC - 1);
    const float* xp = x + (size_t)b * (CC * NN) + (size_t)c * NN;
    int t = threadIdx.x;
    float s = 0.f, sq = 0.f;
    float mx = -__builtin_inff(), mn = __builtin_inff();
    #pragma unroll
    for (int q = t; q < NN; q += 256) {
        float v = xp[q];
        s += v; sq += v * v;
        mx = fmaxf(mx, v); mn = fminf(mn, v);
    }
    __shared__ float ss[256], ssq[256], smx[256], smn[256];
    ss[t] = s; ssq[t] = sq; smx[t] = mx; smn[t] = mn;
    for (int off = 128; off > 0; off >>= 1) {
        __syncthreads();
        if (t < off) {
            ss[t]  += ss[t + off];
            ssq[t] += ssq[t + off];
            smx[t]  = fmaxf(smx[t], smx[t + off]);
            smn[t]  = fminf(smn[t], smn[t + off]);
        }
    }
    if (t == 0) {
        float mean = ss[0] * (1.0f / NN);
        float var  = (ssq[0] - ss[0] * mean) * (1.0f / (NN - 1));
        float sd   = sqrtf(fmaxf(var, 0.f));
        float* pc = pool_c + (size_t)b * (4 * CC) + c;
        pc[0 * CC] = mean; pc[1 * CC] = sd; pc[2 * CC] = mx; pc[3 * CC] = mn;
    }
}

// ---------------------------------------------------------------------------
// Kernel 2: SAME cross-correlation (4 -> 1, odd k, zero pad) + bias + ReLU.
// Produces wgt_t (B,3,N) and wgt_c (B,3,C); j in {0,1,2} maps to k in {3,5,7}.
// ---------------------------------------------------------------------------
__global__ void conv_relu(const float* __restrict__ pool_t, const float* __restrict__ pool_c,
                          const float* __restrict__ w_t3, const float* __restrict__ b_t3,
                          const float* __restrict__ w_c3, const float* __restrict__ b_c3,
                          const float* __restrict__ w_t5, const float* __restrict__ b_t5,
                          const float* __restrict__ w_c5, const float* __restrict__ b_c5,
                          const float* __restrict__ w_t7, const float* __restrict__ b_t7,
                          const float* __restrict__ w_c7, const float* __restrict__ b_c7,
                          float* __restrict__ wgt_t, float* __restrict__ wgt_c) {
    const int totT = BB * 3 * NN;
    const int totC = BB * 3 * CC;
    int gid = blockIdx.x * blockDim.x + threadIdx.x;
    if (gid >= totT + totC) return;
    bool isT = gid < totT;
    int L   = isT ? NN : CC;
    int idx = isT ? gid : gid - totT;
    int l  = idx % L;
    int bj = idx / L;
    int j  = bj % 3;
    int b  = bj / 3;
    const float* pool = (isT ? pool_t : pool_c) + (size_t)b * 4 * L;
    int k = 3 + 2 * j;
    const float* w;
    float bias;
    if (isT) {
        w    = (j == 0) ? w_t3 : (j == 1) ? w_t5 : w_t7;
        bias = ((j == 0) ? b_t3 : (j == 1) ? b_t5 : b_t7)[0];
    } else {
        w    = (j == 0) ? w_c3 : (j == 1) ? w_c5 : w_c7;
        bias = ((j == 0) ? b_c3 : (j == 1) ? b_c5 : b_c7)[0];
    }
    float acc = bias;
    int half = k >> 1;
    for (int i = 0; i < 4; ++i)
        for (int h = 0; h < k; ++h) {
            int p = l + h - half;
            if (p >= 0 && p < L) acc += pool[i * L + p] * w[i * k + h];
        }
    float r = fmaxf(acc, 0.f);
    (isT ? wgt_t : wgt_c)[((size_t)b * 3 + j) * L + l] = r;
}

// ---------------------------------------------------------------------------
// Kernel 3: batched GEMV via V_WMMA_F32_16X16X4_F32 (the only full-fp32 WMMA).
// rows[b, j, m] = sum_k wgt[b, j, k] * x[b*C*N + m*strideM + k*strideK]
// One wave per 16-row M tile. B-matrix columns 0..2 carry the 3 weight
// vectors (k=3,5,7); columns 3..15 are zero.
// Compile-time shapes -> no integer division, dense unrolled WMMA body.
// Two independent accumulator chains hide the WMMA D->C RAW latency.
// ISA layouts (CDNA5 ISA): A/B VGPR v, lane-half h -> K = v + 2h,
// A row M = lane&15, B col = lane&15; D VGPR i -> M = i + 8h, col = lane&15.
// ---------------------------------------------------------------------------
template <int Mdim, int Kdim, int StrideM, int StrideK>
__global__ void gemv_wmma(const float* __restrict__ x, const float* __restrict__ wgt,
                          float* __restrict__ rows) {
    constexpr int mtiles = Mdim >> 4;
    int tile = blockIdx.x & (mtiles - 1);   // mtiles is a power of two
    int b    = blockIdx.x / mtiles;         // folds to a shift
    int lane  = threadIdx.x;                // 0..31 (wave32)
    int mlo   = lane & 15;
    int khalf = lane >> 4;

    const float* xb = x + (size_t)b * (CC * NN);
    const float* wb = wgt + (size_t)b * 3 * Kdim;

    int   j   = mlo;                        // this lane's output column
    int   jj  = (j < 3) ? j : 2;            // clamped: no divergence at WMMA
    float sel = (j < 3) ? 1.0f : 0.0f;
    const float* wrow = wb + (size_t)jj * Kdim;

    const float* arow = xb + (size_t)(tile * 16 + mlo) * StrideM;

    v8f acc0 = {};
    v8f acc1 = {};
    #pragma unroll 4
    for (int k0 = 0; k0 < Kdim; k0 += 8) {
        int ka = k0 + 2 * khalf;
        v2f a0, b0, a1, b1;
        a0.x = arow[(size_t)(ka + 0) * StrideK];
        a0.y = arow[(size_t)(ka + 1) * StrideK];
        a1.x = arow[(size_t)(ka + 4) * StrideK];
        a1.y = arow[(size_t)(ka + 5) * StrideK];
        b0.x = wrow[ka + 0] * sel;
        b0.y = wrow[ka + 1] * sel;
        b1.x = wrow[ka + 4] * sel;
        b1.y = wrow[ka + 5] * sel;
        acc0 = __builtin_amdgcn_wmma_f32_16x16x4_f32(
            false, a0, false, b0, (short)0, acc0, false, false);
        acc1 = __builtin_amdgcn_wmma_f32_16x16x4_f32(
            false, a1, false, b1, (short)0, acc1, false, false);
    }
    if (j < 3) {
        float* o = rows + ((size_t)b * 3 + j) * Mdim + tile * 16 + 8 * khalf;
        #pragma unroll
        for (int i = 0; i < 8; ++i) o[i] = acc0[i] + acc1[i];
    }
}

// ---------------------------------------------------------------------------
// Kernel 4: out[b, n, j*C + c] = 0.5 * (row_t[b,j,c] + row_c[b,j,n]).
// 100 MB of streaming writes -> non-temporal stores (don't pollute L2).
// One 768-thread block per (b, n); fully coalesced.
// ---------------------------------------------------------------------------
__global__ void assemble(const float* __restrict__ row_t, const float* __restrict__ row_c,
                         float* __restrict__ out) {
    int bn = blockIdx.x;
    int b  = bn >> 10;
    int n  = bn & (NN - 1);
    int tid = threadIdx.x;              // 0..767 = j*256 + c
    int j   = tid >> 8;
    float rt = row_t[(size_t)b * (3 * CC) + tid];
    float rc = row_c[((size_t)b * 3 + j) * NN + n];
    float v  = 0.5f * (rt + rc);
    __builtin_nontemporal_store(v, out + (size_t)bn * (3 * CC) + tid);
}

// ---------------------------------------------------------------------------
extern "C" void kernel_launch(void* const* d_in, const int* in_sizes, int n_in,
                              void* d_out, int out_size, void* d_ws, size_t ws_size,
                              hipStream_t stream) {
    const float* x    = (const float*)d_in[0];
    const float* w_t3 = (const float*)d_in[1];
    const float* b_t3 = (const float*)d_in[2];
    const float* w_c3 = (const float*)d_in[3];
    const float* b_c3 = (const float*)d_in[4];
    const float* w_t5 = (const float*)d_in[5];
    const float* b_t5 = (const float*)d_in[6];
    const float* w_c5 = (const float*)d_in[7];
    const float* b_c5 = (const float*)d_in[8];
    const float* w_t7 = (const float*)d_in[9];
    const float* b_t7 = (const float*)d_in[10];
    const float* w_c7 = (const float*)d_in[11];
    const float* b_c7 = (const float*)d_in[12];
    float* out = (float*)d_out;

    // Workspace layout (floats): ~1.6 MB total.
    float* ws     = (float*)d_ws;
    float* pool_t = ws;                               // B*4*N = 131072
    float* pool_c = pool_t + BB * 4 * NN;             // B*4*C =  32768
    float* wgt_t  = pool_c + BB * 4 * CC;             // B*3*N =  98304
    float* wgt_c  = wgt_t  + BB * 3 * NN;             // B*3*C =  24576
    float* row_t  = wgt_c  + BB * 3 * CC;             // B*3*C =  24576
    float* row_c  = row_t  + BB * 3 * CC;             // B*3*N =  98304

    pool_over_C<<<(BB * NN) / 256, 256, 0, stream>>>(x, pool_t);
    pool_over_N<<<BB * CC, 256, 0, stream>>>(x, pool_c);

    int convTot = BB * 3 * (NN + CC);
    conv_relu<<<(convTot + 255) / 256, 256, 0, stream>>>(
        pool_t, pool_c,
        w_t3, b_t3, w_c3, b_c3,
        w_t5, b_t5, w_c5, b_c5,
        w_t7, b_t7, w_c7, b_c7,
        wgt_t, wgt_c);

    // row_t[b,j,c] = sum_n wgt_t[b,j,n] * x[b,c,n]  (M=C, K=N)
    gemv_wmma<CC, NN, NN, 1><<<BB * (CC / 16), 32, 0, stream>>>(x, wgt_t, row_t);
    // row_c[b,j,n] = sum_c wgt_c[b,j,c] * x[b,c,n]  (M=N, K=C)
    gemv_wmma<NN, CC, 1, NN><<<BB * (NN / 16), 32, 0, stream>>>(x, wgt_c, row_c);

    assemble<<<BB * NN, 3 * CC, 0, stream>>>(row_t, row_c, out);
}